// BackboneDistanceEmbedding_13932873908958
// MI455X (gfx1250) — compile-verified
//
#include <hip/hip_runtime.h>
#include <hip/hip_bf16.h>

typedef __attribute__((ext_vector_type(2))) float v2f;
typedef __attribute__((ext_vector_type(8))) float v8f;

#define TILE 16
#define KNN  32

// ---------------------------------------------------------------------------
// Kernel 1: extract packed positions (x,y,z,|p|^2) from affines into ws and
// write the `positions` output.
// ---------------------------------------------------------------------------
__global__ void prep_kernel(const float* __restrict__ aff,
                            float4* __restrict__ pos4,
                            float* __restrict__ pos_out, int n) {
    int i = blockIdx.x * blockDim.x + threadIdx.x;
    if (i >= n) return;
    const float* a = aff + (size_t)i * 16;
    float x = a[3], y = a[7], z = a[11];
    pos4[i] = make_float4(x, y, z, x * x + y * y + z * z);
    pos_out[i * 3 + 0] = x;
    pos_out[i * 3 + 1] = y;
    pos_out[i * 3 + 2] = z;
}

// ---------------------------------------------------------------------------
// Kernel 2: one wave (32 lanes) per 16-row tile. Gram tiles via
// v_wmma_f32_16x16x4_f32, streaming per-lane top-32 kept in LDS.
// ---------------------------------------------------------------------------
__global__ __launch_bounds__(32) void knn_kernel(const float4* __restrict__ pos4,
                                                 float* __restrict__ edge_out,
                                                 int n) {
    __shared__ float dotbuf[TILE * 17];   // 16x16 dot tile, padded stride
    __shared__ float sqc[32];             // |col point|^2 (lanes 16-31: shadow)
    __shared__ float md[TILE * 64];       // per-row: two 32-entry candidate lists
    __shared__ int   mi[TILE * 64];

    const int lane = threadIdx.x;         // 0..31
    const int m    = lane & 15;
    const int hi   = lane >> 4;
    const int r0   = blockIdx.x * TILE;

    // A fragment (rows r0..r0+15): lanes 0-15 hold K=0,1 (x,y); lanes 16-31 K=2,3 (z,0)
    float4 pr = pos4[r0 + m];
    v2f a;
    a.x = hi ? pr.z : pr.x;
    a.y = hi ? 0.0f : pr.y;

    // Selection ownership: 2 lanes per row, 8 columns each
    const int   selRow = lane >> 1;            // 0..15
    const int   selC0  = (lane & 1) * 8;       // 0 or 8
    const int   grow   = r0 + selRow;          // global row for selection
    const float sqRow  = pos4[grow].w;
    const int   base   = selRow * 64 + (lane & 1) * 32;

    for (int i = 0; i < 32; ++i) { md[base + i] = __builtin_inff(); mi[base + i] = 0; }
    float curMax = __builtin_inff();
    int   maxPos = 0;

    const int ntiles = n / TILE;
    for (int t = 0; t < ntiles; ++t) {
        const int c0 = t * TILE;
        float4 pc = pos4[c0 + m];
        // Speculative prefetch of the next column tile (safe OOB per ISA)
        __builtin_prefetch(&pos4[c0 + TILE + m], 0, 3);

        sqc[lane] = pc.w;   // all lanes store: no exec-mask churn in hot loop

        // B fragment (cols c0..c0+15), same half-wave K split as A
        v2f b;
        b.x = hi ? pc.z : pc.x;
        b.y = hi ? 0.0f : pc.y;

        v8f c = {};
        c = __builtin_amdgcn_wmma_f32_16x16x4_f32(
                /*neg_a=*/false, a, /*neg_b=*/false, b,
                /*c_mod=*/(short)0, c, /*reuse_a=*/false, /*reuse_b=*/false);

        // Spill C tile to LDS: lane holds rows (8*hi + r), column m
#pragma unroll
        for (int r = 0; r < 8; ++r)
            dotbuf[(hi * 8 + r) * 17 + m] = c[r];

        __syncthreads();

        // Streaming top-32: threshold check in registers; rare insert rescans LDS
#pragma unroll
        for (int cc = 0; cc < 8; ++cc) {
            const int col  = selC0 + cc;
            const int gcol = c0 + col;
            float d2 = sqRow + sqc[col] - 2.0f * dotbuf[selRow * 17 + col];
            if (gcol != grow && d2 < curMax) {
                md[base + maxPos] = d2;
                mi[base + maxPos] = gcol;
                float mx = md[base];
                int   mp = 0;
                for (int i = 1; i < 32; ++i) {
                    float v = md[base + i];
                    if (v > mx) { mx = v; mp = i; }
                }
                curMax = mx; maxPos = mp;
            }
        }
        __syncthreads();
    }

    // Merge + ascending sort (tie -> lower index, matching top_k), 1 lane per row
    if (lane < TILE) {
        const int rowBase = lane * 64;
        const int growOut = r0 + lane;
        for (int s = 0; s < KNN; ++s) {
            float best = md[rowBase];
            int   bi   = mi[rowBase];
            int   bp   = 0;
            for (int i = 1; i < 64; ++i) {
                float v  = md[rowBase + i];
                int   ix = mi[rowBase + i];
                if (v < best || (v == best && ix < bi)) { best = v; bi = ix; bp = i; }
            }
            md[rowBase + bp] = __builtin_inff();
            mi[rowBase + bp] = 0x7FFFFFFF;
            edge_out[(size_t)growOut * KNN + s] = (float)bi;
        }
    }
}

// ---------------------------------------------------------------------------
// Kernel 3: per-edge neighbor transform + full_edge_index
// out[n,kk,i] = sum_j rot[n,j,i] * (pos[idx]-pos[n])[j]   (R^T * diff)
// ---------------------------------------------------------------------------
__global__ void transform_kernel(const float* __restrict__ aff,
                                 const float4* __restrict__ pos4,
                                 const float* __restrict__ edge,
                                 float* __restrict__ neigh,
                                 float* __restrict__ fe,
                                 int n) {
    int tid = blockIdx.x * blockDim.x + threadIdx.x;
    if (tid >= n * KNN) return;
    int nn = tid >> 5;                 // KNN == 32
    float idxf = edge[tid];
    int   idx  = (int)idxf;

    fe[tid]                     = idxf;        // row 0: source (neighbor) ids
    fe[(size_t)n * KNN + tid]   = (float)nn;   // row 1: targets

    float4 p = pos4[idx];
    float4 q = pos4[nn];
    float dx = p.x - q.x, dy = p.y - q.y, dz = p.z - q.z;
    const float* A = aff + (size_t)nn * 16;
    neigh[(size_t)tid * 3 + 0] = A[0] * dx + A[4] * dy + A[8]  * dz;
    neigh[(size_t)tid * 3 + 1] = A[1] * dx + A[5] * dy + A[9]  * dz;
    neigh[(size_t)tid * 3 + 2] = A[2] * dx + A[6] * dy + A[10] * dz;
}

// ---------------------------------------------------------------------------
extern "C" void kernel_launch(void* const* d_in, const int* in_sizes, int n_in,
                              void* d_out, int out_size, void* d_ws, size_t ws_size,
                              hipStream_t stream) {
    const float* aff = (const float*)d_in[0];
    const int n = in_sizes[0] / 16;            // N x 4 x 4 affines
    float* out = (float*)d_out;

    float* pos_out = out;                                   // N*3
    float* neigh   = out + (size_t)n * 3;                   // N*K*3
    float* edge    = neigh + (size_t)n * KNN * 3;           // N*K
    float* fe      = edge + (size_t)n * KNN;                // 2*N*K

    float4* pos4 = (float4*)d_ws;                           // N * 16B

    prep_kernel<<<(n + 255) / 256, 256, 0, stream>>>(aff, pos4, pos_out, n);
    knn_kernel<<<n / TILE, 32, 0, stream>>>(pos4, edge, n);
    transform_kernel<<<(n * KNN + 255) / 256, 256, 0, stream>>>(aff, pos4, edge, neigh, fe, n);
}